// DeepseekV32MLA_88235808129253
// MI455X (gfx1250) — compile-verified
//
#include <hip/hip_runtime.h>

typedef float v2f __attribute__((ext_vector_type(2)));
typedef float v8f __attribute__((ext_vector_type(8)));
typedef int   v4i __attribute__((vector_size(16)));   // matches async-LDS builtin param

// ---------------- problem constants ----------------
static constexpr int  kB    = 32;
static constexpr int  kS    = 8192;
static constexpr int  kHID  = 7168;
static constexpr int  kH    = 128;
static constexpr int  kQL   = 1536;
static constexpr int  kKVL  = 512;    // + DR(64) = 576 kv width
static constexpr int  kIH   = 64;
static constexpr int  kID   = 128;
static constexpr int  kTOPK = 2048;
static constexpr float kScaleAttn = 0.07216878364870323f;  // (128+64)^-0.5
static constexpr float kScaleIdx  = 0.08838834764831845f;  // 128^-0.5

// ---------------- workspace layout (float offsets) ----------------
static constexpr long alup(long x) { return (x + 1023L) & ~1023L; }
static constexpr long OFF_QKV    = 0;                                     // [32][2240]
static constexpr long OFF_QN     = alup(OFF_QKV    + 32L * 2240);         // [32][1536]
static constexpr long OFF_KVCUR  = alup(OFF_QN     + 32L * 1536);         // [32][576]
static constexpr long OFF_IDXK   = alup(OFF_KVCUR  + 32L * 576);          // [32][128]
static constexpr long OFF_WHEAD  = alup(OFF_IDXK   + 32L * 128);          // [32][64]
static constexpr long OFF_QIDX   = alup(OFF_WHEAD  + 32L * 64);           // [32][64][128]
static constexpr long OFF_LOGIDX = alup(OFF_QIDX   + 32L * 8192);         // [32][64][8192]
static constexpr long OFF_SCORES = alup(OFF_LOGIDX + 32L * 64 * 8192);    // [32][8193]
static constexpr long OFF_TOPK   = alup(OFF_SCORES + 32L * 8193);         // int [32][2048]
static constexpr long OFF_QRAW   = alup(OFF_TOPK   + 32L * 2048);         // [32][128][192]
static constexpr long OFF_Q      = alup(OFF_QRAW   + 32L * 24576);        // [32][128][576]
static constexpr long OFF_ATTN   = alup(OFF_Q      + 32L * 73728);        // [32][128][2048]
static constexpr long OFF_O1     = alup(OFF_ATTN   + 32L * 262144);       // [32][128][512]
static constexpr long OFF_O2     = alup(OFF_O1     + 32L * 65536);        // [32][16384]
static constexpr long OFF_PART   = alup(OFF_O2     + 32L * 16384);        // split-K partials

// ---------------- async global->LDS (CDNA5) ----------------
#if defined(__has_builtin)
#  if __has_builtin(__builtin_amdgcn_global_load_async_to_lds_b128)
#    define HAVE_ASYNC_LDS 1
#  endif
#endif

template <int CPOL>
__device__ __forceinline__ void async_copy16(const float* gsrc, float* ldst)
{
#if defined(HAVE_ASYNC_LDS)
    __builtin_amdgcn_global_load_async_to_lds_b128(
        (__attribute__((address_space(1))) v4i*)gsrc,
        (__attribute__((address_space(3))) v4i*)ldst,
        0, CPOL);
#else
    *(float4*)ldst = *(const float4*)gsrc;
#endif
}

__device__ __forceinline__ void wait_async_all()
{
#if defined(HAVE_ASYNC_LDS)
#  if __has_builtin(__builtin_amdgcn_s_wait_asynccnt)
    __builtin_amdgcn_s_wait_asynccnt(0);
#  else
    asm volatile("s_wait_asynccnt 0" ::: "memory");
#  endif
#endif
}

// =====================================================================
// Generic skinny GEMM, fp32 via V_WMMA_F32_16X16X4_F32, double-buffered
// async-to-LDS staging, specialized by template:
//   TRANSB=0: B row-major [K,N]; LDS layout Bs[k][n] (async-contiguous)
//   TRANSB=1: B row-major [N,K]; LDS layout Bs[n][k] (async-contiguous)
//   GATHER : B row r -> gidx[r]; index == altRow reads the Balt row.
//   BCPOL  : cache policy for the B stream (1 = non-temporal).
// Block = 256 threads = 8 wave32, each wave owns one 16x16 tile of a
// 32x64 output block. Requires M%32==0, N%64==0, (K/ksplit)%32==0.
// =====================================================================
#define AS_(buf, m, k)  AsR[(buf) * 1152 + (m) * 36 + (k)]     // 32 x 36
#define BT_(buf, n, k)  BsR[(buf) * 2304 + (n) * 36 + (k)]     // 64 x 36
#define BN_(buf, k, n)  BsR[(buf) * 2176 + (k) * 68 + (n)]     // 32 x 68

template <int TRANSB, int GATHER, int BCPOL>
__global__ __launch_bounds__(256) void gemm_f32_wmma(
    const float* __restrict__ A, long lda, long sAb,
    const float* __restrict__ Bm, long ldb, long sBb,
    float* __restrict__ C, long ldc, long sCb,
    int K,
    const int* __restrict__ gidx, long sIdx,
    const float* __restrict__ Balt, long sAltB, int altRow,
    float alpha, int ksplit, long sKsplit)
{
    __shared__ float AsR[2 * 32 * 36];
    __shared__ float BsR[2 * 64 * 36];

    const int tid  = threadIdx.x;
    const int lane = tid & 31;
    const int wv   = tid >> 5;
    const int mt   = wv >> 2;                 // 0..1  (m tile)
    const int nt   = wv & 3;                  // 0..3  (n tile)
    const int nblk = blockIdx.x * 64;
    const int mblk = blockIdx.y * 32;
    const int zb   = blockIdx.z;
    const int batch = zb / ksplit;
    const int ks    = zb - batch * ksplit;
    const int Ks    = K / ksplit;
    const int kbeg  = ks * Ks;
    const int nchunks = Ks >> 5;

    A  += (long)batch * sAb;
    Bm += (long)batch * sBb;
    C  += (long)batch * sCb + (long)ks * sKsplit;
    const int*   gi   = GATHER ? (gidx + (long)batch * sIdx)  : nullptr;
    const float* balt = GATHER ? (Balt + (long)batch * sAltB) : nullptr;

    // staging coordinates
    const int arow = tid >> 3;            // 0..31
    const int ac4  = (tid & 7) << 2;      // 0..28
    const float* aRow = A + (long)(mblk + arow) * lda;

    // transB: row pointers (incl. gather / alt resolution) are K-invariant
    const float* tRow0 = nullptr;
    const float* tRow1 = nullptr;
    if constexpr (TRANSB) {
        int nr0 = nblk + arow, nr1 = nr0 + 32;
        if constexpr (GATHER) {
            int g0 = gi[nr0], g1 = gi[nr1];
            tRow0 = (g0 == altRow) ? balt : (Bm + (long)g0 * ldb);
            tRow1 = (g1 == altRow) ? balt : (Bm + (long)g1 * ldb);
        } else {
            tRow0 = Bm + (long)nr0 * ldb;
            tRow1 = Bm + (long)nr1 * ldb;
        }
    }

    // fragment coordinates
    const int am = (mt << 4) + (lane & 15);
    const int bn = (nt << 4) + (lane & 15);
    const int kk = (lane >> 4) << 1;      // lanes 0-15 -> K pair 0,1 ; 16-31 -> 2,3

    v8f acc = {0.f, 0.f, 0.f, 0.f, 0.f, 0.f, 0.f, 0.f};

    // ---- stage one chunk (async issue) ----
    auto stage = [&](int buf, int k0) {
        async_copy16<0>(aRow + k0 + ac4, &AS_(buf, arow, ac4));
        if constexpr (TRANSB) {
            async_copy16<BCPOL>(tRow0 + k0 + ac4, &BT_(buf, arow, ac4));
            async_copy16<BCPOL>(tRow1 + k0 + ac4, &BT_(buf, arow + 32, ac4));
            if constexpr (!GATHER)
                __builtin_prefetch((const void*)(tRow0 + k0 + 64), 0, 1);
        } else {
#pragma unroll
            for (int it = 0; it < 2; ++it) {
                int idx = tid + it * 256;        // 0..511
                int kr  = idx >> 4;              // 0..31
                int c4  = (idx & 15) << 2;       // 0..60
                const float* src;
                if constexpr (GATHER) {
                    int g = gi[k0 + kr];
                    src = (g == altRow) ? (balt + nblk + c4)
                                        : (Bm + (long)g * ldb + nblk + c4);
                } else {
                    src = Bm + (long)(k0 + kr) * ldb + nblk + c4;
                    __builtin_prefetch((const void*)(src + 64 * ldb), 0, 1);
                }
                async_copy16<BCPOL>(src, &BN_(buf, kr, c4));
            }
        }
    };

    stage(0, kbeg);
    wait_async_all();
    __syncthreads();

    for (int c = 0; c < nchunks; ++c) {
        const int cur = c & 1;
        if (c + 1 < nchunks) stage(cur ^ 1, kbeg + ((c + 1) << 5));

        // ---- 8 x V_WMMA_F32_16X16X4_F32 on current buffer ----
#pragma unroll
        for (int k4 = 0; k4 < 32; k4 += 4) {
            v2f a = *(const v2f*)&AS_(cur, am, k4 + kk);
            v2f b;
            if constexpr (TRANSB) {
                b = *(const v2f*)&BT_(cur, bn, k4 + kk);
            } else {
                b.x = BN_(cur, k4 + kk, bn);
                b.y = BN_(cur, k4 + kk + 1, bn);
            }
            acc = __builtin_amdgcn_wmma_f32_16x16x4_f32(
                false, a, false, b, (short)0, acc, false, false);
        }
        wait_async_all();
        __syncthreads();
    }

    // ---- store D (C/D layout: vgpr r, lanes<16 -> m=r, lanes>=16 -> m=r+8) ----
    {
        const int n  = nblk + (nt << 4) + (lane & 15);
        const int m0 = mblk + (mt << 4) + ((lane >> 4) << 3);
#pragma unroll
        for (int r = 0; r < 8; ++r)
            C[(long)(m0 + r) * ldc + n] = acc[r] * alpha;
    }
}

// ---------------- split-K partial reduction ----------------
__global__ void reduce_parts(float* __restrict__ dst, const float* __restrict__ parts,
                             long n, int p, long stride)
{
    long i = (long)blockIdx.x * blockDim.x + threadIdx.x;
    if (i < n) {
        float a = 0.f;
        for (int j = 0; j < p; ++j) a += parts[(long)j * stride + i];
        dst[i] = a;
    }
}

// ---------------- norms / rope / w_head prep (one block per batch) ----------------
__device__ __forceinline__ float block_sum(float v, float* red)
{
    int tid = threadIdx.x;
    red[tid] = v; __syncthreads();
    for (int o = 128; o > 0; o >>= 1) {
        if (tid < o) red[tid] += red[tid + o];
        __syncthreads();
    }
    float r = red[0]; __syncthreads();
    return r;
}

__global__ __launch_bounds__(256) void prep_kernel(
    const float* __restrict__ qkv, const float* __restrict__ hidden,
    const int* __restrict__ positions,
    const float* __restrict__ g_q, const float* __restrict__ g_kv,
    const float* __restrict__ g_idxk, const float* __restrict__ Ww,
    float* __restrict__ q_n, float* __restrict__ kv_cur,
    float* __restrict__ idxk_cur, float* __restrict__ w_head)
{
    __shared__ float red[256];
    __shared__ float tmp[128];
    const int b = blockIdx.x, tid = threadIdx.x;
    const float* row = qkv + (long)b * 2240;
    const float pos = (float)positions[b];
    const float lgt = logf(10000.f);

    // RMS(q_lora[1536])
    float ss = 0.f;
    for (int i = tid; i < 1536; i += 256) { float x = row[i]; ss += x * x; }
    float r = rsqrtf(block_sum(ss, red) / 1536.f + 1e-6f);
    for (int i = tid; i < 1536; i += 256) q_n[(long)b * 1536 + i] = row[i] * r * g_q[i];

    // RMS(k_nope[512]) -> kv_cur[0:512]
    ss = 0.f;
    for (int i = tid; i < 512; i += 256) { float x = row[1664 + i]; ss += x * x; }
    r = rsqrtf(block_sum(ss, red) / 512.f + 1e-6f);
    for (int i = tid; i < 512; i += 256)
        kv_cur[(long)b * 576 + i] = row[1664 + i] * r * g_kv[i];

    // rope(k_pe[64]) -> kv_cur[512:576]
    if (tid < 32) {
        int i = tid;
        float inv = __expf(-lgt * (float)i / 32.f);
        float ang = pos * inv, c = __cosf(ang), s = __sinf(ang);
        float x1 = row[2176 + 2 * i], x2 = row[2176 + 2 * i + 1];
        kv_cur[(long)b * 576 + 512 + 2 * i]     = x1 * c - x2 * s;
        kv_cur[(long)b * 576 + 512 + 2 * i + 1] = x2 * c + x1 * s;
    }

    // LN(index_k[128]) + rope last 64 -> idxk_cur
    float sm = 0.f;
    for (int i = tid; i < 128; i += 256) sm += row[1536 + i];
    float mean = block_sum(sm, red) / 128.f;
    float vv = 0.f;
    for (int i = tid; i < 128; i += 256) { float d = row[1536 + i] - mean; vv += d * d; }
    r = rsqrtf(block_sum(vv, red) / 128.f + 1e-6f);
    if (tid < 128) tmp[tid] = (row[1536 + tid] - mean) * r * g_idxk[tid];
    __syncthreads();
    if (tid < 64) idxk_cur[(long)b * 128 + tid] = tmp[tid];
    if (tid < 32) {
        int i = tid;
        float inv = __expf(-lgt * (float)i / 32.f);
        float ang = pos * inv, c = __cosf(ang), s = __sinf(ang);
        float x1 = tmp[64 + 2 * i], x2 = tmp[64 + 2 * i + 1];
        idxk_cur[(long)b * 128 + 64 + 2 * i]     = x1 * c - x2 * s;
        idxk_cur[(long)b * 128 + 64 + 2 * i + 1] = x2 * c + x1 * s;
    }
    __syncthreads();

    // w_head = (hidden @ Ww) * 64^-0.5
    {
        int j = tid & 63, kp = tid >> 6;
        float acc = 0.f;
        for (int k = kp; k < kHID; k += 4)
            acc += hidden[(long)b * kHID + k] * Ww[(long)k * 64 + j];
        red[tid] = acc; __syncthreads();
        if (tid < 64)
            w_head[(long)b * 64 + tid] =
                (red[tid] + red[tid + 64] + red[tid + 128] + red[tid + 192]) * 0.125f;
    }
}

// in-place rope of elements [off, off+64) of each (b, row); grid (B, rows), 32 thr
__global__ void rope64_kernel(float* __restrict__ x, const int* __restrict__ positions,
                              long rowsPerB, long ld, long off)
{
    int b = blockIdx.x, h = blockIdx.y, i = threadIdx.x;
    float pos = (float)positions[b];
    float* p = x + ((long)b * rowsPerB + h) * ld + off;
    float inv = __expf(-logf(10000.f) * (float)i / 32.f);
    float ang = pos * inv, c = __cosf(ang), s = __sinf(ang);
    float x1 = p[2 * i], x2 = p[2 * i + 1];
    p[2 * i]     = x1 * c - x2 * s;
    p[2 * i + 1] = x2 * c + x1 * s;
}

// rope(q_pe) from q_raw[:, :, 128:192] into Q[:, :, 512:576]; grid (32,128), 32 thr
__global__ void qpe_rope_kernel(const float* __restrict__ q_raw, float* __restrict__ Q,
                                const int* __restrict__ positions)
{
    int b = blockIdx.x, h = blockIdx.y, i = threadIdx.x;
    float pos = (float)positions[b];
    const float* src = q_raw + ((long)b * 128 + h) * 192 + 128;
    float* dst = Q + ((long)b * 128 + h) * 576 + 512;
    float inv = __expf(-logf(10000.f) * (float)i / 32.f);
    float ang = pos * inv, c = __cosf(ang), s = __sinf(ang);
    float x1 = src[2 * i], x2 = src[2 * i + 1];
    dst[2 * i]     = x1 * c - x2 * s;
    dst[2 * i + 1] = x2 * c + x1 * s;
}

// scores[b,s] = sum_h w_head[b,h] * relu(logits[b,h,s]); special col s=8192 inline
__global__ __launch_bounds__(256) void scores_kernel(
    const float* __restrict__ logits, const float* __restrict__ w_head,
    const float* __restrict__ q_idx, const float* __restrict__ idxk_cur,
    float* __restrict__ scores)
{
    __shared__ float wh[64];
    __shared__ float red[256];
    int b = blockIdx.x, y = blockIdx.y, tid = threadIdx.x;
    if (tid < 64) wh[tid] = w_head[(long)b * 64 + tid];
    __syncthreads();
    if (y < 32) {
        int s = y * 256 + tid;
        const float* lg = logits + (long)b * 64 * 8192 + s;
        float acc = 0.f;
#pragma unroll 8
        for (int h = 0; h < 64; ++h)
            acc += wh[h] * fmaxf(lg[(long)h * 8192], 0.f);
        scores[(long)b * 8193 + s] = acc;
    } else {
        float val = 0.f;
        if (tid < 64) {
            const float* q = q_idx + (long)b * 8192 + (long)tid * 128;
            const float* kk = idxk_cur + (long)b * 128;
            float d = 0.f;
            for (int i = 0; i < 128; ++i) d += q[i] * kk[i];
            val = wh[tid] * fmaxf(d * kScaleIdx, 0.f);
        }
        red[tid] = val; __syncthreads();
        for (int o = 128; o > 0; o >>= 1) {
            if (tid < o) red[tid] += red[tid + o];
            __syncthreads();
        }
        if (tid == 0) scores[(long)b * 8193 + 8192] = red[0];
    }
}

// deterministic per-batch top-2048 of 8193 scores: 4-round radix select on
// order-preserving uint keys, then prefix-scan compaction (no global atomics)
__device__ __forceinline__ unsigned score_key(float f)
{
    unsigned u = __float_as_uint(f);
    return (u & 0x80000000u) ? ~u : (u | 0x80000000u);
}

__global__ __launch_bounds__(256) void topk_kernel(const float* __restrict__ scores,
                                                   int* __restrict__ topk)
{
    const int NS = 8193, K = kTOPK;
    int b = blockIdx.x, tid = threadIdx.x;
    const float* sc = scores + (long)b * NS;
    __shared__ int hist[256];
    __shared__ int scan[256];
    __shared__ unsigned shPrefix;
    __shared__ int shNeed;
    if (tid == 0) { shPrefix = 0u; shNeed = K; }
    __syncthreads();
    unsigned prefmask = 0u;
    for (int round = 0; round < 4; ++round) {
        int shift = 24 - 8 * round;
        hist[tid] = 0;
        __syncthreads();
        unsigned prefix = shPrefix;
        for (int s = tid; s < NS; s += 256) {
            unsigned u = score_key(sc[s]);
            if ((u & prefmask) == prefix) atomicAdd(&hist[(u >> shift) & 255], 1);
        }
        __syncthreads();
        if (tid == 0) {
            int need = shNeed, acc = 0, v = 255;
            for (; v > 0; --v) {
                if (acc + hist[v] >= need) break;
                acc += hist[v];
            }
            shNeed = need - acc;
            shPrefix = prefix | ((unsigned)v << shift);
        }
        prefmask |= 0xFFu << shift;
        __syncthreads();
    }
    const unsigned T = shPrefix;
    const int need = shNeed;

    // pass 1: strictly greater than threshold
    int cnt = 0;
    for (int s = tid; s < NS; s += 256)
        if (score_key(sc[s]) > T) ++cnt;
    scan[tid] = cnt; __syncthreads();
    for (int o = 1; o < 256; o <<= 1) {
        int v = (tid >= o) ? scan[tid - o] : 0;
        __syncthreads();
        scan[tid] += v;
        __syncthreads();
    }
    int base = (tid == 0) ? 0 : scan[tid - 1];
    int totalGt = scan[255];
    __syncthreads();
    int p = base;
    for (int s = tid; s < NS; s += 256)
        if (score_key(sc[s]) > T) topk[(long)b * K + (p++)] = s;

    // pass 2: equal to threshold, take exactly `need`
    cnt = 0;
    for (int s = tid; s < NS; s += 256)
        if (score_key(sc[s]) == T) ++cnt;
    __syncthreads();
    scan[tid] = cnt; __syncthreads();
    for (int o = 1; o < 256; o <<= 1) {
        int v = (tid >= o) ? scan[tid - o] : 0;
        __syncthreads();
        scan[tid] += v;
        __syncthreads();
    }
    base = (tid == 0) ? 0 : scan[tid - 1];
    __syncthreads();
    p = totalGt + base;
    for (int s = tid; s < NS && p < K; s += 256)
        if (score_key(sc[s]) == T) { topk[(long)b * K + p] = s; ++p; }
}

// row softmax over 2048, in place; one block per (b, head)
__global__ __launch_bounds__(256) void softmax2048(float* __restrict__ x)
{
    __shared__ float red[256];
    float* p = x + (long)blockIdx.x * 2048;
    int tid = threadIdx.x;
    float m = -1e30f;
    for (int i = tid; i < 2048; i += 256) m = fmaxf(m, p[i]);
    red[tid] = m; __syncthreads();
    for (int o = 128; o > 0; o >>= 1) {
        if (tid < o) red[tid] = fmaxf(red[tid], red[tid + o]);
        __syncthreads();
    }
    m = red[0]; __syncthreads();
    float s = 0.f;
    for (int i = tid; i < 2048; i += 256) { float e = __expf(p[i] - m); p[i] = e; s += e; }
    red[tid] = s; __syncthreads();
    for (int o = 128; o > 0; o >>= 1) {
        if (tid < o) red[tid] += red[tid + o];
        __syncthreads();
    }
    float inv = 1.f / red[0];
    for (int i = tid; i < 2048; i += 256) p[i] *= inv;
}

// ---------------- host-side launch ----------------
static void launch_gemm(hipStream_t st,
    const float* A, long lda, long sAb,
    const float* Bm, long ldb, long sBb, int transB,
    float* C, long ldc, long sCb,
    int M, int N, int K,
    const int* gidx, long sIdx,
    const float* Balt, long sAltB, int altRow,
    float alpha, int batch, int ksplit, long sKsplit)
{
    dim3 grid(N / 64, M / 32, batch * ksplit);
    dim3 blk(256);
    if (transB) {
        if (gidx)
            gemm_f32_wmma<1, 1, 0><<<grid, blk, 0, st>>>(A, lda, sAb, Bm, ldb, sBb,
                C, ldc, sCb, K, gidx, sIdx, Balt, sAltB, altRow, alpha, ksplit, sKsplit);
        else
            gemm_f32_wmma<1, 0, 1><<<grid, blk, 0, st>>>(A, lda, sAb, Bm, ldb, sBb,
                C, ldc, sCb, K, gidx, sIdx, Balt, sAltB, altRow, alpha, ksplit, sKsplit);
    } else {
        if (gidx)
            gemm_f32_wmma<0, 1, 0><<<grid, blk, 0, st>>>(A, lda, sAb, Bm, ldb, sBb,
                C, ldc, sCb, K, gidx, sIdx, Balt, sAltB, altRow, alpha, ksplit, sKsplit);
        else
            gemm_f32_wmma<0, 0, 1><<<grid, blk, 0, st>>>(A, lda, sAb, Bm, ldb, sBb,
                C, ldc, sCb, K, gidx, sIdx, Balt, sAltB, altRow, alpha, ksplit, sKsplit);
    }
}

extern "C" void kernel_launch(void* const* d_in, const int* in_sizes, int n_in,
                              void* d_out, int out_size, void* d_ws, size_t ws_size,
                              hipStream_t stream)
{
    (void)in_sizes; (void)n_in; (void)out_size; (void)ws_size;
    const float* hidden = (const float*)d_in[0];
    const int*   pos    = (const int*)d_in[1];
    const float* kvc    = (const float*)d_in[2];   // [32][8192][576]
    const float* ikc    = (const float*)d_in[3];   // [32][8192][128]
    const float* W_a    = (const float*)d_in[4];
    const float* g_q    = (const float*)d_in[5];
    const float* g_kv   = (const float*)d_in[6];
    const float* W_qb   = (const float*)d_in[7];
    const float* w_kc   = (const float*)d_in[8];
    const float* w_vc   = (const float*)d_in[9];
    const float* W_o    = (const float*)d_in[10];
    const float* Wq_idx = (const float*)d_in[11];
    const float* Ww     = (const float*)d_in[12];
    const float* g_idxk = (const float*)d_in[13];
    float* out = (float*)d_out;
    float* w   = (float*)d_ws;

    float* qkv      = w + OFF_QKV;
    float* q_n      = w + OFF_QN;
    float* kv_cur   = w + OFF_KVCUR;
    float* idxk_cur = w + OFF_IDXK;
    float* w_head   = w + OFF_WHEAD;
    float* q_idx    = w + OFF_QIDX;
    float* logidx   = w + OFF_LOGIDX;
    float* scores   = w + OFF_SCORES;
    int*   topk     = (int*)(w + OFF_TOPK);
    float* q_raw    = w + OFF_QRAW;
    float* Qb       = w + OFF_Q;
    float* attn     = w + OFF_ATTN;
    float* o1       = w + OFF_O1;
    float* o2       = w + OFF_O2;
    float* parts    = w + OFF_PART;

    // 1) qkv = hidden @ W_a   (split-K x8 for occupancy, then reduce)
    launch_gemm(stream, hidden, kHID, 0, W_a, 2240, 0, 0, parts, 2240, 0,
                32, 2240, kHID, nullptr, 0, nullptr, 0, 0, 1.f, 1, 8, 32L * 2240);
    reduce_parts<<<(32 * 2240 + 255) / 256, 256, 0, stream>>>(qkv, parts, 32L * 2240, 8, 32L * 2240);

    // 2) norms, ropes, w_head
    prep_kernel<<<32, 256, 0, stream>>>(qkv, hidden, pos, g_q, g_kv, g_idxk, Ww,
                                        q_n, kv_cur, idxk_cur, w_head);

    // 3) q_idx = q_n @ Wq_idx
    launch_gemm(stream, q_n, kQL, 0, Wq_idx, 8192, 0, 0, q_idx, 8192, 0,
                32, 8192, kQL, nullptr, 0, nullptr, 0, 0, 1.f, 1, 1, 0);
    // 4) rope q_idx last 64 dims per head
    rope64_kernel<<<dim3(32, 64), 32, 0, stream>>>(q_idx, pos, 64, 128, 64);

    // 5) indexer logits = q_idx @ index_k_cache^T  (batched over b, scaled ID^-0.5)
    launch_gemm(stream, q_idx, 128, 8192, ikc, 128, (long)kS * kID, 1,
                logidx, 8192, 64L * 8192, kIH, 8192, 128,
                nullptr, 0, nullptr, 0, 0, kScaleIdx, kB, 1, 0);

    // 6) scores (+ s==8192 column inline)
    scores_kernel<<<dim3(32, 33), 256, 0, stream>>>(logidx, w_head, q_idx, idxk_cur, scores);

    // 7) deterministic top-2048
    topk_kernel<<<32, 256, 0, stream>>>(scores, topk);

    // 8) q_raw = q_n @ W_qb
    launch_gemm(stream, q_n, kQL, 0, W_qb, 24576, 0, 0, q_raw, 24576, 0,
                32, 24576, kQL, nullptr, 0, nullptr, 0, 0, 1.f, 1, 1, 0);
    // 9) rope(q_pe) -> Q[:, :, 512:576]
    qpe_rope_kernel<<<dim3(32, 128), 32, 0, stream>>>(q_raw, Qb, pos);

    // 10) q_abs = q_nope @ w_kc[h] -> Q[:, :, 0:512]  (batched over 128 heads)
    launch_gemm(stream, q_raw, 24576, 192, w_kc, 512, 65536, 0, Qb, 73728, 576,
                32, 512, 128, nullptr, 0, nullptr, 0, 0, 1.f, kH, 1, 0);

    // 11) attn logits = Q @ kv_sel^T (gather through topk, alt row = kv_cur)
    launch_gemm(stream, Qb, 576, 73728, kvc, 576, (long)kS * 576, 1,
                attn, 2048, 262144, kH, kTOPK, 576,
                topk, kTOPK, kv_cur, 576, kS, kScaleAttn, kB, 1, 0);

    // 12) softmax over selected 2048
    softmax2048<<<32 * 128, 256, 0, stream>>>(attn);

    // 13) o1 = attn @ kv_sel[:, :512] (gathered rows again, hot in L2)
    launch_gemm(stream, attn, 2048, 262144, kvc, 576, (long)kS * 576, 0,
                o1, 512, 65536, kH, kKVL, kTOPK,
                topk, kTOPK, kv_cur, 576, kS, 1.f, kB, 1, 0);

    // 14) o2 = o1 @ w_vc[h]  (batched over heads, packed to [32][16384])
    launch_gemm(stream, o1, 65536, 512, w_vc, 128, 65536, 0, o2, 16384, 128,
                32, 128, kKVL, nullptr, 0, nullptr, 0, 0, 1.f, kH, 1, 0);

    // 15) out = o2 @ W_o  (split-K x2, then reduce into d_out)
    launch_gemm(stream, o2, 16384, 0, W_o, kHID, 0, 0, parts, kHID, 0,
                32, kHID, 16384, nullptr, 0, nullptr, 0, 0, 1.f, 1, 2, 32L * kHID);
    reduce_parts<<<(32 * kHID + 255) / 256, 256, 0, stream>>>(out, parts, 32L * kHID, 2, 32L * kHID);
}